// ResidualDownBlock_60172491817015
// MI455X (gfx1250) — compile-verified
//
#include <hip/hip_runtime.h>
#include <hip/hip_bf16.h>

typedef __attribute__((ext_vector_type(2))) float v2f;
typedef __attribute__((ext_vector_type(8))) float v8f;

#define CIN 64
#define COUT 128
#define LDSA_STRIDE 72   // 64 + pad; 72*4=288 bytes => float4-aligned rows, spreads banks

// Async gather of one 16-byte chunk straight into LDS (gfx1250 ASYNCcnt path).
__device__ __forceinline__ void async_gather_b128(const float* src, float* ldsDst) {
    unsigned lds = (unsigned)(size_t)ldsDst;                 // flat->LDS offset (low 32 bits)
    unsigned long long ga = (unsigned long long)(size_t)src;
    asm volatile("global_load_async_to_lds_b128 %0, %1, off"
                 :: "v"(lds), "v"(ga) : "memory");
}
__device__ __forceinline__ void wait_async() {
    asm volatile("s_wait_asynccnt 0x0" ::: "memory");
}

// ---------------------------------------------------------------------------
// Conv gather-GEMM-scatter: out_acc[out_idx[k][slot]] += feats[in_idx[k][slot]] @ Wc[k]
// grid = (ceil(P/32), 9), block = 256 (8 waves; wave w covers cols [16w,16w+16)).
// Each wave keeps TWO 16x16 accumulators (rows tile..+15 and tile+16..+31) that
// share one B fragment per k-step: 32 v_wmma_f32_16x16x4_f32 per wave.
// ---------------------------------------------------------------------------
__global__ __launch_bounds__(256) void conv_wmma_kernel(
    const float* __restrict__ feats, const float* __restrict__ Wc,
    const int* __restrict__ in_idx, const int* __restrict__ out_idx,
    float* __restrict__ out_acc, int N, int P)
{
    __shared__ float ldsA[32 * LDSA_STRIDE];
    const int tap  = blockIdx.y;
    const int tile = blockIdx.x * 32;
    const int t    = threadIdx.x;

    // Stage 32x64 gathered A tile into LDS: 512 float4 chunks, 2 per thread.
#pragma unroll
    for (int i = t; i < 32 * 16; i += 256) {
        int r = i >> 4, seg = i & 15;
        int slot = tile + r;
        int ridx = (slot < P) ? in_idx[(long)tap * P + slot] : N;
        float* dst = &ldsA[r * LDSA_STRIDE + seg * 4];
        if (ridx < N) {
            async_gather_b128(feats + (long)ridx * CIN + seg * 4, dst);
        } else {
            *(float4*)dst = make_float4(0.f, 0.f, 0.f, 0.f);
        }
    }
    wait_async();
    __syncthreads();

    const int lane = t & 31;
    const int col0 = (t >> 5) * 16;
    const int n    = lane & 15;   // A row in tile / B,D col in tile
    const int hi   = lane >> 4;   // K half select

    v8f acc0 = {}, acc1 = {};
    const float* wbase = Wc + (long)tap * CIN * COUT + col0 + n;

#pragma unroll
    for (int ks = 0; ks < CIN / 4; ++ks) {
        const int kb = ks * 4 + hi * 2;
        v2f a0, a1, b;
        a0.x = ldsA[n * LDSA_STRIDE + kb + 0];
        a0.y = ldsA[n * LDSA_STRIDE + kb + 1];
        a1.x = ldsA[(n + 16) * LDSA_STRIDE + kb + 0];
        a1.y = ldsA[(n + 16) * LDSA_STRIDE + kb + 1];
        b.x  = wbase[(long)(kb + 0) * COUT];
        b.y  = wbase[(long)(kb + 1) * COUT];
        acc0 = __builtin_amdgcn_wmma_f32_16x16x4_f32(false, a0, false, b,
                                                     (short)0, acc0, false, false);
        acc1 = __builtin_amdgcn_wmma_f32_16x16x4_f32(false, a1, false, b,
                                                     (short)0, acc1, false, false);
    }

    // Scatter: D VGPR v holds D[m = v + 8*hi][n]
#pragma unroll
    for (int v = 0; v < 8; ++v) {
        int m = v + hi * 8;
        int slot0 = tile + m;
        int slot1 = tile + 16 + m;
        if (slot0 < P) {
            int orow = out_idx[(long)tap * P + slot0];
            atomicAdd(&out_acc[(long)orow * COUT + col0 + n], acc0[v]);
        }
        if (slot1 < P) {
            int orow = out_idx[(long)tap * P + slot1];
            atomicAdd(&out_acc[(long)orow * COUT + col0 + n], acc1[v]);
        }
    }
}

// ---------------------------------------------------------------------------
// Skip GEMM: skip_feats[slot] = feats[skip_in_idx[slot]] @ Ws   (direct store)
// Same 32-row double-accumulator tiling.
// ---------------------------------------------------------------------------
__global__ __launch_bounds__(256) void skip_wmma_kernel(
    const float* __restrict__ feats, const float* __restrict__ Ws,
    const int* __restrict__ in_idx, float* __restrict__ skip_feats,
    int N, int S)
{
    __shared__ float ldsA[32 * LDSA_STRIDE];
    const int tile = blockIdx.x * 32;
    const int t    = threadIdx.x;
#pragma unroll
    for (int i = t; i < 32 * 16; i += 256) {
        int r = i >> 4, seg = i & 15;
        int slot = tile + r;
        int ridx = (slot < S) ? in_idx[slot] : N;
        float* dst = &ldsA[r * LDSA_STRIDE + seg * 4];
        if (ridx < N) {
            async_gather_b128(feats + (long)ridx * CIN + seg * 4, dst);
        } else {
            *(float4*)dst = make_float4(0.f, 0.f, 0.f, 0.f);
        }
    }
    wait_async();
    __syncthreads();

    const int lane = t & 31;
    const int col0 = (t >> 5) * 16;
    const int n    = lane & 15;
    const int hi   = lane >> 4;

    v8f acc0 = {}, acc1 = {};
    const float* wbase = Ws + col0 + n;
#pragma unroll
    for (int ks = 0; ks < CIN / 4; ++ks) {
        const int kb = ks * 4 + hi * 2;
        v2f a0, a1, b;
        a0.x = ldsA[n * LDSA_STRIDE + kb + 0];
        a0.y = ldsA[n * LDSA_STRIDE + kb + 1];
        a1.x = ldsA[(n + 16) * LDSA_STRIDE + kb + 0];
        a1.y = ldsA[(n + 16) * LDSA_STRIDE + kb + 1];
        b.x  = wbase[(long)(kb + 0) * COUT];
        b.y  = wbase[(long)(kb + 1) * COUT];
        acc0 = __builtin_amdgcn_wmma_f32_16x16x4_f32(false, a0, false, b,
                                                     (short)0, acc0, false, false);
        acc1 = __builtin_amdgcn_wmma_f32_16x16x4_f32(false, a1, false, b,
                                                     (short)0, acc1, false, false);
    }
#pragma unroll
    for (int v = 0; v < 8; ++v) {
        int m = v + hi * 8;
        int slot0 = tile + m;
        int slot1 = tile + 16 + m;
        if (slot0 < S) skip_feats[(long)slot0 * COUT + col0 + n] = acc0[v];
        if (slot1 < S) skip_feats[(long)slot1 * COUT + col0 + n] = acc1[v];
    }
}

// ---------------------------------------------------------------------------
// Column sum / sum-of-squares (128 threads, thread == channel)
// ---------------------------------------------------------------------------
#define STAT_CHUNK 2048L
__global__ __launch_bounds__(128) void colstats_kernel(
    const float* __restrict__ X, int rows, float* __restrict__ stats)
{
    const int c = threadIdx.x;
    long r0 = (long)blockIdx.x * STAT_CHUNK;
    long r1 = r0 + STAT_CHUNK; if (r1 > rows) r1 = rows;
    float s = 0.f, s2 = 0.f;
    for (long r = r0; r < r1; ++r) {
        float v = X[r * COUT + c];
        s += v; s2 += v * v;
    }
    atomicAdd(&stats[c], s);
    atomicAdd(&stats[COUT + c], s2);
}

__global__ __launch_bounds__(256) void bnfinal_kernel(
    const float* __restrict__ convStats, const float* __restrict__ skipStats,
    float* __restrict__ convMuInv, float* __restrict__ skipMuInv, int M, int S)
{
    int t = threadIdx.x;
    if (t < COUT) {
        float mu  = convStats[t] / (float)M;
        float var = convStats[COUT + t] / (float)M - mu * mu;
        convMuInv[t] = mu;
        convMuInv[COUT + t] = rsqrtf(var + 1e-5f);
    } else {
        int c = t - COUT;
        float mu  = skipStats[c] / (float)S;
        float var = skipStats[COUT + c] / (float)S - mu * mu;
        skipMuInv[c] = mu;
        skipMuInv[COUT + c] = rsqrtf(var + 1e-5f);
    }
}

// ---------------------------------------------------------------------------
// Per-batch segment sums of BN(out) (thread == channel; register accumulators)
// ---------------------------------------------------------------------------
__global__ __launch_bounds__(128) void segsum_kernel(
    const float* __restrict__ out_acc, const int* __restrict__ batch_of_out,
    const float* __restrict__ muinv, const float* __restrict__ gamma,
    const float* __restrict__ beta, float* __restrict__ seg,
    float* __restrict__ cnt, int M)
{
    const int c = threadIdx.x;
    long r0 = (long)blockIdx.x * STAT_CHUNK;
    long r1 = r0 + STAT_CHUNK; if (r1 > M) r1 = M;
    const float mu = muinv[c], inv = muinv[COUT + c], g = gamma[c], bt = beta[c];
    float s0 = 0.f, s1 = 0.f;
    int c0 = 0, c1 = 0;
    for (long r = r0; r < r1; ++r) {
        int b = batch_of_out[r];
        float v = (out_acc[r * COUT + c] - mu) * inv * g + bt;
        if (b == 0) { s0 += v; c0++; } else { s1 += v; c1++; }
    }
    atomicAdd(&seg[c], s0);
    atomicAdd(&seg[COUT + c], s1);
    if (c == 0) { atomicAdd(&cnt[0], (float)c0); atomicAdd(&cnt[1], (float)c1); }
}

// ---------------------------------------------------------------------------
// Squeeze-excite FC: desc(2x128) -> relu(fc1) -> sigmoid(fc2) = attn(2x128)
// ---------------------------------------------------------------------------
__global__ __launch_bounds__(128) void se_kernel(
    const float* __restrict__ seg, const float* __restrict__ cnt,
    const float* __restrict__ fc1, const float* __restrict__ fc2,
    float* __restrict__ attn)
{
    __shared__ float desc[2 * COUT];
    __shared__ float h[16];
    const int t = threadIdx.x;
    desc[t]        = seg[t]        / cnt[0];
    desc[COUT + t] = seg[COUT + t] / cnt[1];
    __syncthreads();
    if (t < 16) {
        int b = t >> 3, j = t & 7;
        float s = 0.f;
        for (int c = 0; c < COUT; ++c) s += desc[b * COUT + c] * fc1[c * 8 + j];
        h[t] = fmaxf(s, 0.f);
    }
    __syncthreads();
    for (int b = 0; b < 2; ++b) {
        float s = 0.f;
#pragma unroll
        for (int j = 0; j < 8; ++j) s += h[b * 8 + j] * fc2[j * COUT + t];
        attn[b * COUT + t] = 1.f / (1.f + expf(-s));
    }
}

// ---------------------------------------------------------------------------
// out = BN(out_acc) * attn[batch]   -- float4 streaming, full overwrite of d_out
// ---------------------------------------------------------------------------
__global__ __launch_bounds__(256) void scale_kernel(
    const float* __restrict__ out_acc, const int* __restrict__ batch_of_out,
    const float* __restrict__ muinv, const float* __restrict__ gamma,
    const float* __restrict__ beta, const float* __restrict__ attn,
    float* __restrict__ out, long total4)
{
    long i4 = (long)blockIdx.x * blockDim.x + threadIdx.x;
    if (i4 >= total4) return;
    long i  = i4 << 2;
    int  c0 = (int)(i & (COUT - 1));
    long r  = i >> 7;
    int  b  = batch_of_out[r];
    float4 v  = *(const float4*)(out_acc + i);
    float4 mu = *(const float4*)(muinv + c0);
    float4 iv = *(const float4*)(muinv + COUT + c0);
    float4 g  = *(const float4*)(gamma + c0);
    float4 bt = *(const float4*)(beta + c0);
    float4 at = *(const float4*)(attn + b * COUT + c0);
    float4 o;
    o.x = ((v.x - mu.x) * iv.x * g.x + bt.x) * at.x;
    o.y = ((v.y - mu.y) * iv.y * g.y + bt.y) * at.y;
    o.z = ((v.z - mu.z) * iv.z * g.z + bt.z) * at.z;
    o.w = ((v.w - mu.w) * iv.w * g.w + bt.w) * at.w;
    *(float4*)(out + i) = o;
}

__global__ __launch_bounds__(256) void skipadd_kernel(
    const float* __restrict__ skip_feats, const int* __restrict__ skip_out_idx,
    const float* __restrict__ muinv, const float* __restrict__ gamma,
    const float* __restrict__ beta, float* __restrict__ out, long total4)
{
    long i4 = (long)blockIdx.x * blockDim.x + threadIdx.x;
    if (i4 >= total4) return;
    long i  = i4 << 2;
    int  c0 = (int)(i & (COUT - 1));
    long s  = i >> 7;
    long r  = skip_out_idx[s];
    float4 v  = *(const float4*)(skip_feats + i);
    float4 mu = *(const float4*)(muinv + c0);
    float4 iv = *(const float4*)(muinv + COUT + c0);
    float4 g  = *(const float4*)(gamma + c0);
    float4 bt = *(const float4*)(beta + c0);
    float* dst = out + r * COUT + c0;          // rows unique -> no atomics
    float4 d = *(const float4*)dst;
    d.x += (v.x - mu.x) * iv.x * g.x + bt.x;
    d.y += (v.y - mu.y) * iv.y * g.y + bt.y;
    d.z += (v.z - mu.z) * iv.z * g.z + bt.z;
    d.w += (v.w - mu.w) * iv.w * g.w + bt.w;
    *(float4*)dst = d;
}

__global__ __launch_bounds__(256) void relu_kernel(float* __restrict__ out, long total4)
{
    long i4 = (long)blockIdx.x * blockDim.x + threadIdx.x;
    if (i4 >= total4) return;
    float4* p = (float4*)out + i4;
    float4 v = *p;
    v.x = fmaxf(v.x, 0.f); v.y = fmaxf(v.y, 0.f);
    v.z = fmaxf(v.z, 0.f); v.w = fmaxf(v.w, 0.f);
    *p = v;
}

// ---------------------------------------------------------------------------
extern "C" void kernel_launch(void* const* d_in, const int* in_sizes, int n_in,
                              void* d_out, int out_size, void* d_ws, size_t ws_size,
                              hipStream_t stream) {
    const float* feats    = (const float*)d_in[0];
    const float* W_conv   = (const float*)d_in[1];
    const float* W_skip   = (const float*)d_in[2];
    const float* bn_g     = (const float*)d_in[3];
    const float* bn_b     = (const float*)d_in[4];
    const float* sbn_g    = (const float*)d_in[5];
    const float* sbn_b    = (const float*)d_in[6];
    const float* fc1      = (const float*)d_in[7];
    const float* fc2      = (const float*)d_in[8];
    const int* conv_in    = (const int*)d_in[9];
    const int* conv_out   = (const int*)d_in[10];
    const int* skip_in    = (const int*)d_in[11];
    const int* skip_out   = (const int*)d_in[12];
    const int* batch_of   = (const int*)d_in[13];

    const int N = in_sizes[0] / CIN;
    const int P = in_sizes[9] / 9;
    const int S = in_sizes[11];
    const int M = in_sizes[13];

    // Workspace layout (floats)
    float* ws = (float*)d_ws;
    float* out_acc    = ws;                                  // (M+1)*128
    float* skip_feats = out_acc + (size_t)(M + 1) * COUT;    // S*128
    float* convStats  = skip_feats + (size_t)S * COUT;       // 256 (sum, sumsq)
    float* skipStats  = convStats + 2 * COUT;                // 256
    float* convMuInv  = skipStats + 2 * COUT;                // 256 (mu, rsig)
    float* skipMuInv  = convMuInv + 2 * COUT;                // 256
    float* seg        = skipMuInv + 2 * COUT;                // 256
    float* cnt        = seg + 2 * COUT;                      // 2
    float* attn       = cnt + 2;                             // 256

    hipMemsetAsync(out_acc, 0, (size_t)(M + 1) * COUT * sizeof(float), stream);
    hipMemsetAsync(convStats, 0,
                   (size_t)((attn + 2 * COUT) - convStats) * sizeof(float), stream);

    // 1) conv gather-GEMM-scatter (async-to-LDS gather + WMMA fp32 + atomic scatter)
    {
        dim3 grid((P + 31) / 32, 9);
        conv_wmma_kernel<<<grid, 256, 0, stream>>>(feats, W_conv, conv_in, conv_out,
                                                   out_acc, N, P);
    }
    // 2) skip GEMM
    skip_wmma_kernel<<<(S + 31) / 32, 256, 0, stream>>>(feats, W_skip, skip_in,
                                                        skip_feats, N, S);
    // 3) BN stats for both paths
    {
        int nbM = (int)((M + STAT_CHUNK - 1) / STAT_CHUNK);
        int nbS = (int)((S + STAT_CHUNK - 1) / STAT_CHUNK);
        colstats_kernel<<<nbM, 128, 0, stream>>>(out_acc, M, convStats);
        colstats_kernel<<<nbS, 128, 0, stream>>>(skip_feats, S, skipStats);
        bnfinal_kernel<<<1, 256, 0, stream>>>(convStats, skipStats,
                                              convMuInv, skipMuInv, M, S);
    }
    // 4) per-batch segment sums + SE FC
    {
        int nbM = (int)((M + STAT_CHUNK - 1) / STAT_CHUNK);
        segsum_kernel<<<nbM, 128, 0, stream>>>(out_acc, batch_of, convMuInv,
                                               bn_g, bn_b, seg, cnt, M);
        se_kernel<<<1, 128, 0, stream>>>(seg, cnt, fc1, fc2, attn);
    }
    // 5) out = BN(out)*attn ; += BN(skip) scatter ; relu   (float4 streaming)
    {
        long t4M = (long)M * COUT / 4;
        long t4S = (long)S * COUT / 4;
        int gM = (int)((t4M + 255) / 256);
        int gS = (int)((t4S + 255) / 256);
        scale_kernel<<<gM, 256, 0, stream>>>(out_acc, batch_of, convMuInv,
                                             bn_g, bn_b, attn, (float*)d_out, t4M);
        skipadd_kernel<<<gS, 256, 0, stream>>>(skip_feats, skip_out, skipMuInv,
                                               sbn_g, sbn_b, (float*)d_out, t4S);
        relu_kernel<<<gM, 256, 0, stream>>>((float*)d_out, t4M);
    }
}